// DyGNN_36704790511957
// MI455X (gfx1250) — compile-verified
//
#include <hip/hip_runtime.h>
#include <hip/hip_bf16.h>

// ---------------------------------------------------------------------------
// DyGNN streaming recurrence for MI455X (gfx1250), wave32.
//   - One persistent 256-thread (8-wave) workgroup runs the S-step scan.
//   - All matvecs use V_WMMA_F32_16X16X4_F32 (full-f32 matrix pipe) so the
//     2048-step recurrence keeps reference precision.
//   - Node-state updates are logged per step; a parallel pre-pass computes
//     "last prior occurrence" pointers so gathers read log-or-table.
//   - A-tile carrier lanes are selected by mask-multiply (VALU select), not
//     divergent loads, so EXEC stays all-ones around every WMMA.
// Workspace usage: 4*S ints (prev ptrs) + 6*S*128 floats (state logs) ~ 6.3MB.
// ---------------------------------------------------------------------------

typedef float v2f __attribute__((ext_vector_type(2)));
typedef float v8f __attribute__((ext_vector_type(8)));

#define DN 128          // D = E = 128
#define DN4 512         // 4*D
#define NWAVES 8
#define BLOCK 256

__device__ __forceinline__ float sigm(float x) { return 1.0f / (1.0f + expf(-x)); }

// ---------------------------------------------------------------------------
// Wave-level matvec tile: computes, for one 16-wide output chunk [n0, n0+16):
//   y[n] = act( bias[n] + addin[n] + (acc?y[n]:0) + sum_k x1[k]*W1[k][n] (+ x2/W2) )
// via V_WMMA_F32_16X16X4_F32 with the vector in row 0 of the A tile.
// A 16x4 f32 layout: lanes 0-15 hold (K=0,K=1), lanes 16-31 hold (K=2,K=3);
// only lanes 0 and 16 carry row 0, selected by mask multiply (no divergence).
// B 4x16 f32 layout mirrors the K split across lane halves.
// Row 0 of D lands in VGPR0 elem of lanes 0-15.
// ---------------------------------------------------------------------------
template <bool HAS_W2>
__device__ __forceinline__ void mv_tile(
    const float* __restrict__ W1, const float* __restrict__ x1,
    const float* __restrict__ W2, const float* __restrict__ x2,
    int K, int ldW, int n0,
    const float* __restrict__ bias, const float* __restrict__ addin,
    float* __restrict__ y, bool accumulate, bool do_tanh)
{
  const int   lane = threadIdx.x & 31;
  const int   half = lane >> 4;          // 0: K pair {0,1}, 1: K pair {2,3}
  const int   l15  = lane & 15;
  const int   col  = n0 + l15;
  const float msk  = (l15 == 0) ? 1.0f : 0.0f;  // row-0 carrier lanes: 0 and 16

  v8f acc = {};
  for (int k0 = 0; k0 < K; k0 += 4) {
    const int kA = k0 + (half << 1);
    v2f a1, b1;
    a1.x = x1[kA]     * msk;             // LDS broadcast + VALU select
    a1.y = x1[kA + 1] * msk;
    b1.x = W1[(size_t)kA * ldW + col];
    b1.y = W1[(size_t)(kA + 1) * ldW + col];
    acc = __builtin_amdgcn_wmma_f32_16x16x4_f32(false, a1, false, b1,
                                                (short)0, acc, false, false);
    if (HAS_W2) {
      v2f a2, b2;
      a2.x = x2[kA]     * msk;
      a2.y = x2[kA + 1] * msk;
      b2.x = W2[(size_t)kA * ldW + col];
      b2.y = W2[(size_t)(kA + 1) * ldW + col];
      acc = __builtin_amdgcn_wmma_f32_16x16x4_f32(false, a2, false, b2,
                                                  (short)0, acc, false, false);
    }
  }
  if (lane < 16) {
    float r = acc[0];                    // D row 0, N = col
    if (bias)  r += bias[col];
    if (addin) r += addin[col];
    if (accumulate) r += y[col];
    y[col] = do_tanh ? tanhf(r) : r;
  }
}

// ---------------------------------------------------------------------------
// Pre-pass: for each step s, find the last prior step where this step's head
// node / tail node appeared as a head and as a tail (-1 if never).
// ---------------------------------------------------------------------------
__global__ void dygnn_prev_kernel(const int* __restrict__ heads,
                                  const int* __restrict__ tails, int S,
                                  int* __restrict__ prevHH, int* __restrict__ prevHT,
                                  int* __restrict__ prevTH, int* __restrict__ prevTT)
{
  int s = blockIdx.x * blockDim.x + threadIdx.x;
  if (s >= S) return;
  const int h = heads[s], t = tails[s];
  int phh = -1, pht = -1, pth = -1, ptt = -1;
  for (int j = s - 1; j >= 0; --j) {
    const int hj = heads[j], tj = tails[j];
    if (phh < 0 && hj == h) phh = j;
    if (pht < 0 && tj == h) pht = j;
    if (pth < 0 && hj == t) pth = j;
    if (ptt < 0 && tj == t) ptt = j;
    if (phh >= 0 && pht >= 0 && pth >= 0 && ptt >= 0) break;
  }
  prevHH[s] = phh; prevHT[s] = pht; prevTH[s] = pth; prevTT[s] = ptt;
}

// ---------------------------------------------------------------------------
// Main sequential kernel: one workgroup, loops over all S steps.
// ---------------------------------------------------------------------------
__global__ __launch_bounds__(BLOCK) void dygnn_scan_kernel(
    const int* __restrict__ heads, const int* __restrict__ tails,
    const float* __restrict__ times,
    const float* __restrict__ node_rep, const float* __restrict__ cell_head,
    const float* __restrict__ hidden_head, const float* __restrict__ cell_tail,
    const float* __restrict__ hidden_tail,
    const float* __restrict__ Weh1, const float* __restrict__ Weh2, const float* __restrict__ beh,
    const float* __restrict__ Wet1, const float* __restrict__ Wet2, const float* __restrict__ bet,
    const float* __restrict__ Wxh, const float* __restrict__ Whh, const float* __restrict__ bh,
    const float* __restrict__ Wdh, const float* __restrict__ bdh,
    const float* __restrict__ Wxt, const float* __restrict__ Wht, const float* __restrict__ bt,
    const float* __restrict__ Wdt, const float* __restrict__ bdt,
    const float* __restrict__ Wc1, const float* __restrict__ Wc2,
    const int* __restrict__ prevHH, const int* __restrict__ prevHT,
    const int* __restrict__ prevTH, const int* __restrict__ prevTT,
    float* __restrict__ repHlog, float* __restrict__ repTlog,
    float* __restrict__ cHlog, float* __restrict__ hHlog,
    float* __restrict__ cTlog, float* __restrict__ hTlog,
    float* __restrict__ out, int S)
{
  __shared__ float repH[DN], repT[DN];
  __shared__ float chH[DN], hhH[DN], ctT[DN], htT[DN];
  __shared__ float htOldH[DN], hhOldT[DN];      // ht[h_i] / hh[t_i] (pre-step)
  __shared__ float eH[DN], eT[DN];              // edge MLP outputs
  __shared__ float csH[DN], csT[DN];            // tanh(c @ Wd + bd)
  __shared__ float zH[DN4], zT[DN4];            // gate pre-activations
  __shared__ float combPH[DN], combPT[DN];      // combiner partials (old hiddens)
  __shared__ float hNewH[DN], hNewT[DN];
  __shared__ float newRepH[DN], newRepT[DN];
  __shared__ float sDecH, sDecT;

  const int tid  = threadIdx.x;
  const int wave = tid >> 5;

  for (int s = 0; s < S; ++s) {
    const int   h   = heads[s];
    const int   t   = tails[s];
    const float tm  = times[s];
    const int   phh = prevHH[s], pht = prevHT[s];
    const int   pth = prevTH[s], ptt = prevTT[s];

    // ---- gather node state (log if previously touched, else input tables) ----
    for (int i = tid; i < DN; i += BLOCK) {
      // rep: latest of head-role / tail-role writes, tail wins on same step
      float rh = (pht >= phh)
                   ? (pht >= 0 ? repTlog[(size_t)pht * DN + i] : node_rep[(size_t)h * DN + i])
                   :  repHlog[(size_t)phh * DN + i];
      float rt = (ptt >= pth)
                   ? (ptt >= 0 ? repTlog[(size_t)ptt * DN + i] : node_rep[(size_t)t * DN + i])
                   :  repHlog[(size_t)pth * DN + i];
      repH[i] = rh;  repT[i] = rt;
      // if_updated=0: outputs are the pre-update representations
      out[(size_t)s * DN + i]       = rh;
      out[(size_t)(S + s) * DN + i] = rt;

      chH[i]    = (phh >= 0) ? cHlog[(size_t)phh * DN + i] : cell_head  [(size_t)h * DN + i];
      hhH[i]    = (phh >= 0) ? hHlog[(size_t)phh * DN + i] : hidden_head[(size_t)h * DN + i];
      ctT[i]    = (ptt >= 0) ? cTlog[(size_t)ptt * DN + i] : cell_tail  [(size_t)t * DN + i];
      htT[i]    = (ptt >= 0) ? hTlog[(size_t)ptt * DN + i] : hidden_tail[(size_t)t * DN + i];
      htOldH[i] = (pht >= 0) ? hTlog[(size_t)pht * DN + i] : hidden_tail[(size_t)h * DN + i];
      hhOldT[i] = (pth >= 0) ? hHlog[(size_t)pth * DN + i] : hidden_head[(size_t)t * DN + i];
    }
    if (tid == 0) {
      const int pH = (phh > pht) ? phh : pht;
      const int pT = (pth > ptt) ? pth : ptt;
      const float rtH = (pH >= 0) ? times[pH] : 0.0f;
      const float rtT = (pT >= 0) ? times[pT] : 0.0f;
      sDecH = expf(-(tm - rtH));   // W_DECAY = 1.0, exp decay
      sDecT = expf(-(tm - rtT));
    }
    __syncthreads();

    // ---- phase 1: everything computable from step-start state (112 tiles) ----
    for (int u = wave; u < 112; u += NWAVES) {
      if      (u <   8) mv_tile<true >(Weh1, repH, Weh2, repT, DN, DN,  u        * 16, beh, nullptr, eH,  false, true);
      else if (u <  16) mv_tile<true >(Wet1, repH, Wet2, repT, DN, DN, (u -   8) * 16, bet, nullptr, eT,  false, true);
      else if (u <  24) mv_tile<false>(Wdh,  chH,  nullptr, nullptr, DN, DN,  (u - 16) * 16, bdh, nullptr, csH, false, true);
      else if (u <  32) mv_tile<false>(Wdt,  ctT,  nullptr, nullptr, DN, DN,  (u - 24) * 16, bdt, nullptr, csT, false, true);
      else if (u <  64) mv_tile<false>(Whh,  hhH,  nullptr, nullptr, DN, DN4, (u - 32) * 16, bh,  nullptr, zH,  false, false);
      else if (u <  96) mv_tile<false>(Wht,  htT,  nullptr, nullptr, DN, DN4, (u - 64) * 16, bt,  nullptr, zT,  false, false);
      else if (u < 104) mv_tile<false>(Wc2,  htOldH, nullptr, nullptr, DN, DN, (u -  96) * 16, nullptr, nullptr, combPH, false, false);
      else              mv_tile<false>(Wc1,  hhOldT, nullptr, nullptr, DN, DN, (u - 104) * 16, nullptr, nullptr, combPT, false, false);
    }
    __syncthreads();

    // ---- phase 2a: z += edge @ Wx (64 tiles) ----
    for (int u = wave; u < 64; u += NWAVES) {
      if (u < 32) mv_tile<false>(Wxh, eH, nullptr, nullptr, DN, DN4,  u       * 16, nullptr, nullptr, zH, true, false);
      else        mv_tile<false>(Wxt, eT, nullptr, nullptr, DN, DN4, (u - 32) * 16, nullptr, nullptr, zT, true, false);
    }
    __syncthreads();

    // ---- phase 2b: TLSTM gates (elementwise) + write c/h logs ----
    for (int i = tid; i < DN; i += BLOCK) {
      // head TLSTM
      {
        float ig = sigm(zH[i]);
        float fg = sigm(zH[DN + i]);
        float og = sigm(zH[2 * DN + i]);
        float gg = tanhf(zH[3 * DN + i]);
        float cadj = chH[i] - csH[i] + csH[i] * sDecH;
        float c  = fg * cadj + ig * gg;
        float hn = og * tanhf(c);
        cHlog[(size_t)s * DN + i] = c;
        hHlog[(size_t)s * DN + i] = hn;
        hNewH[i] = hn;
      }
      // tail TLSTM
      {
        float ig = sigm(zT[i]);
        float fg = sigm(zT[DN + i]);
        float og = sigm(zT[2 * DN + i]);
        float gg = tanhf(zT[3 * DN + i]);
        float cadj = ctT[i] - csT[i] + csT[i] * sDecT;
        float c  = fg * cadj + ig * gg;
        float hn = og * tanhf(c);
        cTlog[(size_t)s * DN + i] = c;
        hTlog[(size_t)s * DN + i] = hn;
        hNewT[i] = hn;
      }
    }
    __syncthreads();

    // ---- phase 3: combiner (16 tiles) ----
    for (int u = wave; u < 16; u += NWAVES) {
      if (u < 8) mv_tile<false>(Wc1, hNewH, nullptr, nullptr, DN, DN,  u      * 16, nullptr, combPH, newRepH, false, true);
      else       mv_tile<false>(Wc2, hNewT, nullptr, nullptr, DN, DN, (u - 8) * 16, nullptr, combPT, newRepT, false, true);
    }
    __syncthreads();

    // ---- write rep logs, make all logs visible before next iteration ----
    for (int i = tid; i < DN; i += BLOCK) {
      repHlog[(size_t)s * DN + i] = newRepH[i];
      repTlog[(size_t)s * DN + i] = newRepT[i];
    }
    __threadfence_block();
    __syncthreads();
  }
}

// ---------------------------------------------------------------------------
// Host-side launch
// ---------------------------------------------------------------------------
extern "C" void kernel_launch(void* const* d_in, const int* in_sizes, int n_in,
                              void* d_out, int out_size, void* d_ws, size_t ws_size,
                              hipStream_t stream) {
  const int S = in_sizes[0];   // number of streaming interactions

  const int*   heads       = (const int*)  d_in[0];
  const int*   tails       = (const int*)  d_in[1];
  const float* times       = (const float*)d_in[2];
  const float* node_rep    = (const float*)d_in[3];
  const float* cell_head   = (const float*)d_in[4];
  const float* hidden_head = (const float*)d_in[5];
  const float* cell_tail   = (const float*)d_in[6];
  const float* hidden_tail = (const float*)d_in[7];
  const float* Weh1 = (const float*)d_in[8];
  const float* Weh2 = (const float*)d_in[9];
  const float* beh  = (const float*)d_in[10];
  const float* Wet1 = (const float*)d_in[11];
  const float* Wet2 = (const float*)d_in[12];
  const float* bet  = (const float*)d_in[13];
  const float* Wxh  = (const float*)d_in[14];
  const float* Whh  = (const float*)d_in[15];
  const float* bh   = (const float*)d_in[16];
  const float* Wdh  = (const float*)d_in[17];
  const float* bdh  = (const float*)d_in[18];
  const float* Wxt  = (const float*)d_in[19];
  const float* Wht  = (const float*)d_in[20];
  const float* bt   = (const float*)d_in[21];
  const float* Wdt  = (const float*)d_in[22];
  const float* bdt  = (const float*)d_in[23];
  const float* Wc1  = (const float*)d_in[24];
  const float* Wc2  = (const float*)d_in[25];

  // Workspace carve-up: 4*S ints + 6*S*128 floats (~6.3 MB for S=2048)
  char* ws = (char*)d_ws;
  int* prevHH = (int*)ws; ws += (size_t)S * sizeof(int);
  int* prevHT = (int*)ws; ws += (size_t)S * sizeof(int);
  int* prevTH = (int*)ws; ws += (size_t)S * sizeof(int);
  int* prevTT = (int*)ws; ws += (size_t)S * sizeof(int);
  float* repHlog = (float*)ws; ws += (size_t)S * DN * sizeof(float);
  float* repTlog = (float*)ws; ws += (size_t)S * DN * sizeof(float);
  float* cHlog   = (float*)ws; ws += (size_t)S * DN * sizeof(float);
  float* hHlog   = (float*)ws; ws += (size_t)S * DN * sizeof(float);
  float* cTlog   = (float*)ws; ws += (size_t)S * DN * sizeof(float);
  float* hTlog   = (float*)ws; ws += (size_t)S * DN * sizeof(float);

  dygnn_prev_kernel<<<(S + BLOCK - 1) / BLOCK, BLOCK, 0, stream>>>(
      heads, tails, S, prevHH, prevHT, prevTH, prevTT);

  dygnn_scan_kernel<<<1, BLOCK, 0, stream>>>(
      heads, tails, times,
      node_rep, cell_head, hidden_head, cell_tail, hidden_tail,
      Weh1, Weh2, beh, Wet1, Wet2, bet,
      Wxh, Whh, bh, Wdh, bdh,
      Wxt, Wht, bt, Wdt, bdt,
      Wc1, Wc2,
      prevHH, prevHT, prevTH, prevTT,
      repHlog, repTlog, cHlog, hHlog, cTlog, hTlog,
      (float*)d_out, S);
}